// ARGCN_37400575213597
// MI455X (gfx1250) — compile-verified
//
#include <hip/hip_runtime.h>
#include <hip/hip_bf16.h>

#define N_NODES 50000
#define N_GRAPHS 500
#define BLOCKSZ 100
#define N_EDGES 800000
#define DD 32
#define HH 4
#define RR 16
#define LL 2
#define SLOPE 0.2f
#define LAM 0.5f

#define NBLK_E (N_EDGES / 256)          // 3125, exact
#define TILES_MAX (N_EDGES / 16 + RR)   // worst-case 16-edge tiles over all relations

typedef float v2f __attribute__((ext_vector_type(2)));
typedef float v8f __attribute__((ext_vector_type(8)));

__device__ __forceinline__ float lrelu(float x) { return x > 0.f ? x : SLOPE * x; }

// ---------------- h = code_emb[node_ids] ----------------
__global__ void argcn_gather_h(const float* __restrict__ code_emb,
                               const int* __restrict__ node_ids,
                               float* __restrict__ h) {
    int t = blockIdx.x * blockDim.x + threadIdx.x;
    if (t >= N_NODES * DD) return;
    int n = t >> 5, d = t & 31;
    h[t] = code_emb[node_ids[n] * DD + d];
}

// ---------------- per-layer init: agg=0, den=0, emax=-inf ----------------
__global__ void argcn_init_layer(float* __restrict__ agg,
                                 float* __restrict__ den,
                                 float* __restrict__ emax) {
    int t = blockIdx.x * blockDim.x + threadIdx.x;
    if (t < N_NODES * DD) agg[t] = 0.f;
    if (t < N_NODES * HH) { den[t] = 0.f; emax[t] = -3.4e38f; }
}

// ================= relation counting sort (once; rel is layer-invariant) ============
__global__ void argcn_sort_count(const int* __restrict__ rel,
                                 int* __restrict__ blockHist) {
    __shared__ int hist[RR];
    int b = blockIdx.x, tid = threadIdx.x;
    if (tid < RR) hist[tid] = 0;
    __syncthreads();
    atomicAdd(&hist[rel[b * 256 + tid]], 1);
    __syncthreads();
    if (tid < RR) blockHist[b * RR + tid] = hist[tid];
}

// single block of 256: totals, prefix bases, tile bases, per-block scatter offsets
__global__ void argcn_sort_scan(const int* __restrict__ blockHist,
                                int* __restrict__ base,      // [RR+1]
                                int* __restrict__ tileBase,  // [RR+1]
                                int* __restrict__ blockOfs) {// [NBLK_E*RR]
    __shared__ int part[256];
    __shared__ int cnt_s[RR];
    __shared__ int base_s[RR];
    int tid = threadIdx.x;
    int r = tid & 15, chunk = tid >> 4;
    int sum = 0;
    for (int b = chunk; b < NBLK_E; b += 16) sum += blockHist[b * RR + r];
    part[tid] = sum;
    __syncthreads();
    if (tid < RR) {
        int tot = 0;
        for (int c = 0; c < 16; c++) tot += part[c * RR + tid];
        cnt_s[tid] = tot;
    }
    __syncthreads();
    if (tid == 0) {
        int acc = 0, tacc = 0;
        for (int q = 0; q < RR; q++) {
            base[q] = acc; tileBase[q] = tacc; base_s[q] = acc;
            acc += cnt_s[q];
            tacc += (cnt_s[q] + 15) >> 4;
        }
        base[RR] = acc; tileBase[RR] = tacc;
    }
    __syncthreads();
    if (tid < RR) {
        int run = base_s[tid];
        for (int b = 0; b < NBLK_E; b++) {
            blockOfs[b * RR + tid] = run;
            run += blockHist[b * RR + tid];
        }
    }
}

__global__ void argcn_sort_scatter(const int* __restrict__ rel,
                                   const int* __restrict__ blockOfs,
                                   int* __restrict__ perm) {
    __shared__ int cur[RR];
    int b = blockIdx.x, tid = threadIdx.x;
    if (tid < RR) cur[tid] = blockOfs[b * RR + tid];
    __syncthreads();
    int e = b * 256 + tid;
    int pos = atomicAdd(&cur[rel[e]], 1);
    perm[pos] = e;
}

// ---------------- er[r,h] = sum_d attn_r[h,d]*(rel_table@Wr + br)[r,h,d] ----------------
__global__ void argcn_rel_er(const float* __restrict__ rel_table,
                             const float* __restrict__ Wr_l,
                             const float* __restrict__ br_l,
                             const float* __restrict__ attn_r_l,
                             float* __restrict__ er) {
    int t = threadIdx.x;
    if (t >= RR * HH) return;
    int r = t >> 2, hh = t & 3;
    float acc = 0.f;
    for (int d = 0; d < DD; d++) {
        int j = hh * DD + d;
        float sp = br_l[j];
        for (int k = 0; k < DD; k++) sp += rel_table[r * DD + k] * Wr_l[k * (HH * DD) + j];
        acc += attn_r_l[j] * sp;
    }
    er[r * HH + hh] = acc;
}

// ---------------- per graph: es -> ra (softmax over R), esg ----------------
__global__ void argcn_graph_att(const float* __restrict__ state,
                                const float* __restrict__ Wr_l,
                                const float* __restrict__ br_l,
                                const float* __restrict__ We_l,
                                const float* __restrict__ be_l,
                                const float* __restrict__ attn_rs_l,
                                const float* __restrict__ attn_s_l,
                                const float* __restrict__ er,
                                float* __restrict__ ra,    // [G,R,H]
                                float* __restrict__ esg) { // [G,H]
    int g = blockIdx.x;
    int j = threadIdx.x;            // 0..127, head = j/32 matches wave32 boundaries
    int hh = j >> 5, d = j & 31;
    __shared__ float s_st[DD];
    __shared__ float s_es[HH];
    if (j < DD) s_st[j] = state[g * DD + j];
    __syncthreads();
    float sp = br_l[j];
    float gp = be_l[j];
    for (int k = 0; k < DD; k++) {
        float sv = s_st[k];
        sp += sv * Wr_l[k * (HH * DD) + j];
        gp += sv * We_l[k * (HH * DD) + j];
    }
    float vr = sp * attn_rs_l[j];
    float vs = gp * attn_s_l[j];
    for (int off = 16; off > 0; off >>= 1) {
        vr += __shfl_down(vr, off, 32);
        vs += __shfl_down(vs, off, 32);
    }
    if (d == 0) { s_es[hh] = vr; esg[g * HH + hh] = vs; }
    __syncthreads();
    if (j < HH) {
        float es = s_es[j];
        float vals[RR];
        float mx = -3.4e38f;
        for (int r = 0; r < RR; r++) {
            float v = lrelu(er[r * HH + j] + es);
            vals[r] = v; mx = fmaxf(mx, v);
        }
        float sum = 0.f;
        for (int r = 0; r < RR; r++) { float e = expf(vals[r] - mx); vals[r] = e; sum += e; }
        float inv = 1.f / sum;
        for (int r = 0; r < RR; r++) ra[(g * RR + r) * HH + j] = vals[r] * inv;
    }
}

// ---------------- hp = h @ We_l + be_l via V_WMMA_F32_16X16X4_F32 ----------------
__global__ void argcn_hp_gemm(const float* __restrict__ h,
                              const float* __restrict__ We_l,
                              const float* __restrict__ be_l,
                              float* __restrict__ hp) {
    int wave = threadIdx.x >> 5;
    int lane = threadIdx.x & 31;
    int half = lane >> 4;
    int lr   = lane & 15;
    int mtile = blockIdx.x;
    int ncol  = wave * 16 + lr;

    float bias = be_l[ncol];
    v8f c;
    #pragma unroll
    for (int i = 0; i < 8; i++) c[i] = bias;

    const float* arow = h + (mtile * 16 + lr) * DD;
    #pragma unroll
    for (int k = 0; k < 8; k++) {
        v2f a, b;
        a.x = arow[4 * k + 2 * half + 0];
        a.y = arow[4 * k + 2 * half + 1];
        b.x = We_l[(4 * k + 2 * half + 0) * (HH * DD) + ncol];
        b.y = We_l[(4 * k + 2 * half + 1) * (HH * DD) + ncol];
        c = __builtin_amdgcn_wmma_f32_16x16x4_f32(false, a, false, b, (short)0, c,
                                                  false, false);
    }
    #pragma unroll
    for (int i = 0; i < 8; i++) {
        int m = i + 8 * half;
        hp[(mtile * 16 + m) * (HH * DD) + ncol] = c[i];
    }
}

// ---------------- per node: a = esn + eh, b = et ----------------
__global__ void argcn_node_att(const float* __restrict__ hp,
                               const float* __restrict__ attn_h_l,
                               const float* __restrict__ attn_t_l,
                               const float* __restrict__ esg,
                               const int* __restrict__ n2g,
                               float* __restrict__ a_nh,
                               float* __restrict__ b_nh) {
    int t = blockIdx.x * blockDim.x + threadIdx.x;
    if (t >= N_NODES * HH) return;
    int n = t >> 2, hh = t & 3;
    const float* row = hp + n * (HH * DD) + hh * DD;
    float eh = 0.f, et = 0.f;
    for (int d = 0; d < DD; d++) {
        float v = row[d];
        eh += v * attn_h_l[hh * DD + d];
        et += v * attn_t_l[hh * DD + d];
    }
    a_nh[t] = eh + esg[n2g[n] * HH + hh];
    b_nh[t] = et;
}

// ---------------- edge pass 1: e = lrelu(a[src]+b[dst]); segment max ----------------
__global__ void argcn_edge_pass1(const int* __restrict__ src,
                                 const int* __restrict__ dst,
                                 const float* __restrict__ a_nh,
                                 const float* __restrict__ b_nh,
                                 float* __restrict__ ebuf,
                                 float* __restrict__ emax) {
    int t = blockIdx.x * blockDim.x + threadIdx.x;
    if (t >= N_EDGES * HH) return;
    int e = t >> 2, hh = t & 3;
    float v = lrelu(a_nh[src[e] * HH + hh] + b_nh[dst[e] * HH + hh]);
    ebuf[t] = v;
    atomicMax(emax + dst[e] * HH + hh, v);
}

// ---------------- edge pass 2: ex = exp(e - emax[dst]); segment sum ----------------
__global__ void argcn_edge_pass2(const int* __restrict__ dst,
                                 const float* __restrict__ emax,
                                 float* __restrict__ ebuf,
                                 float* __restrict__ den) {
    int t = blockIdx.x * blockDim.x + threadIdx.x;
    if (t >= N_EDGES * HH) return;
    int e = t >> 2, hh = t & 3;
    int d4 = dst[e] * HH + hh;
    float ex = expf(ebuf[t] - emax[d4]);
    ebuf[t] = ex;
    atomicAdd(den + d4, ex);
}

// ---------------- edge pass 3a: edge_att, rel_att, coeff ----------------
__global__ void argcn_edge_coeff(const int* __restrict__ src,
                                 const int* __restrict__ dst,
                                 const int* __restrict__ rel,
                                 const int* __restrict__ n2g,
                                 const float* __restrict__ ebuf,
                                 const float* __restrict__ den,
                                 const float* __restrict__ ra,
                                 float* __restrict__ out_edge,
                                 float* __restrict__ out_rel,
                                 float* __restrict__ coeff) {
    int t = blockIdx.x * blockDim.x + threadIdx.x;
    if (t >= N_EDGES * HH) return;
    int e = t >> 2, hh = t & 3;
    float ea = ebuf[t] / den[dst[e] * HH + hh];
    float rv = ra[(n2g[src[e]] * RR + rel[e]) * HH + hh];
    out_edge[t] = ea;
    out_rel[t]  = rv;
    coeff[t]    = ea * rv;
}

// ---------------- edge pass 3b: WMMA message GEMM over relation-sorted edges --------
// one wave per tile of 16 edges (all same relation):
//   A[16x32] rows: c_e[K] = sum_h coeff[e,h]*hp[src_e,h,K]   (built on the fly)
//   B = W[rel] (32x32), two 16-col accumulators sharing A -> 16 WMMAs / 16 edges
//   D rows scatter-added into agg[dst_e]
__global__ void argcn_edge_msg_wmma(const int* __restrict__ perm,
                                    const int* __restrict__ base,      // [RR+1]
                                    const int* __restrict__ tileBase,  // [RR+1]
                                    const int* __restrict__ src,
                                    const int* __restrict__ dst,
                                    const float* __restrict__ coeff,
                                    const float* __restrict__ hp,
                                    const float* __restrict__ W_l,
                                    float* __restrict__ agg) {
    int t = blockIdx.x * (blockDim.x >> 5) + (threadIdx.x >> 5);
    int totalTiles = tileBase[RR];
    if (t >= totalTiles) return;                 // wave-uniform exit, EXEC stays full
    int r = 0;
    while (r < RR - 1 && t >= tileBase[r + 1]) r++;
    int p0 = base[r] + (t - tileBase[r]) * 16;
    int nvalid = base[r + 1] - p0;
    nvalid = nvalid > 16 ? 16 : nvalid;

    int lane = threadIdx.x & 31;
    int half = lane >> 4;
    int lr   = lane & 15;                        // edge row handled by this lane
    bool v = lr < nvalid;
    int e  = perm[v ? (p0 + lr) : p0];
    int s  = src[e];
    int dn = dst[e];                             // broadcast to output rows via shfl
    float cf0 = v ? coeff[e * HH + 0] : 0.f;
    float cf1 = v ? coeff[e * HH + 1] : 0.f;
    float cf2 = v ? coeff[e * HH + 2] : 0.f;
    float cf3 = v ? coeff[e * HH + 3] : 0.f;

    const float* hrow = hp + (size_t)s * (HH * DD);
    const float* Wr_  = W_l + r * (DD * DD);
    v8f acc0 = {}, acc1 = {};
    #pragma unroll
    for (int k = 0; k < 8; k++) {
        int K = 4 * k + 2 * half;                // A layout: vgpr0 = K, vgpr1 = K+1
        v2f h0 = *(const v2f*)(hrow + 0 * DD + K);
        v2f h1 = *(const v2f*)(hrow + 1 * DD + K);
        v2f h2 = *(const v2f*)(hrow + 2 * DD + K);
        v2f h3 = *(const v2f*)(hrow + 3 * DD + K);
        v2f a, b0, b1;
        a.x = cf0 * h0.x + cf1 * h1.x + cf2 * h2.x + cf3 * h3.x;
        a.y = cf0 * h0.y + cf1 * h1.y + cf2 * h2.y + cf3 * h3.y;
        b0.x = Wr_[K * DD + lr];       b0.y = Wr_[(K + 1) * DD + lr];
        b1.x = Wr_[K * DD + 16 + lr];  b1.y = Wr_[(K + 1) * DD + 16 + lr];
        acc0 = __builtin_amdgcn_wmma_f32_16x16x4_f32(false, a, false, b0, (short)0, acc0,
                                                     false, false);
        acc1 = __builtin_amdgcn_wmma_f32_16x16x4_f32(false, a, false, b1, (short)0, acc1,
                                                     false, false);
    }
    // C/D layout: vgpr i -> row M = i + 8*half, col N = lr (tile0), 16+lr (tile1)
    #pragma unroll
    for (int i = 0; i < 8; i++) {
        int m = i + 8 * half;
        int dnr = __shfl(dn, m, 32);
        if (m < nvalid) {
            atomicAdd(agg + dnr * DD + lr,       acc0[i]);
            atomicAdd(agg + dnr * DD + 16 + lr,  acc1[i]);
        }
    }
}

// ---------------- h = relu((1-LAM)*(agg+bias) + LAM*h) ----------------
__global__ void argcn_update_h(const float* __restrict__ agg,
                               const float* __restrict__ hbias_l,
                               float* __restrict__ h) {
    int t = blockIdx.x * blockDim.x + threadIdx.x;
    if (t >= N_NODES * DD) return;
    float a = agg[t] + hbias_l[t & 31];
    h[t] = fmaxf(0.f, (1.f - LAM) * a + LAM * h[t]);
}

// ---------------- readout: x[g] = sum_{n in g} h[n]*rw[n] ----------------
__global__ void argcn_readout(const float* __restrict__ h,
                              const float* __restrict__ rw,
                              float* __restrict__ out_x) {
    int g = blockIdx.x;
    int d = threadIdx.x;  // 32 threads
    float acc = 0.f;
    for (int i = 0; i < BLOCKSZ; i++) {
        int n = g * BLOCKSZ + i;
        acc += h[n * DD + d] * rw[n];
    }
    out_x[g * DD + d] = acc;
}

extern "C" void kernel_launch(void* const* d_in, const int* in_sizes, int n_in,
                              void* d_out, int out_size, void* d_ws, size_t ws_size,
                              hipStream_t stream) {
    const float* state   = (const float*)d_in[0];
    const float* rw      = (const float*)d_in[1];
    const float* code_emb= (const float*)d_in[2];
    const float* rel_tab = (const float*)d_in[3];
    const float* W       = (const float*)d_in[4];
    const float* h_bias  = (const float*)d_in[5];
    const float* Wr      = (const float*)d_in[6];
    const float* br      = (const float*)d_in[7];
    const float* We      = (const float*)d_in[8];
    const float* be      = (const float*)d_in[9];
    const float* attn_h  = (const float*)d_in[10];
    const float* attn_t  = (const float*)d_in[11];
    const float* attn_s  = (const float*)d_in[12];
    const float* attn_r  = (const float*)d_in[13];
    const float* attn_rs = (const float*)d_in[14];
    const int* node_ids  = (const int*)d_in[15];
    const int* src       = (const int*)d_in[16];
    const int* dst       = (const int*)d_in[17];
    const int* rel       = (const int*)d_in[18];
    const int* n2g       = (const int*)d_in[19];

    float* ws = (float*)d_ws;
    size_t o = 0;
    float* h     = ws + o; o += (size_t)N_NODES * DD;        // 1.6M
    float* hp    = ws + o; o += (size_t)N_NODES * HH * DD;   // 6.4M
    float* a_nh  = ws + o; o += (size_t)N_NODES * HH;
    float* b_nh  = ws + o; o += (size_t)N_NODES * HH;
    float* ebuf  = ws + o; o += (size_t)N_EDGES * HH;        // 3.2M
    float* emax  = ws + o; o += (size_t)N_NODES * HH;
    float* den   = ws + o; o += (size_t)N_NODES * HH;
    float* agg   = ws + o; o += (size_t)N_NODES * DD;        // 1.6M
    float* ra    = ws + o; o += (size_t)N_GRAPHS * RR * HH;
    float* er    = ws + o; o += (size_t)RR * HH;
    float* esg   = ws + o; o += (size_t)N_GRAPHS * HH;
    float* coeff = ws + o; o += (size_t)N_EDGES * HH;        // 3.2M
    int* wsI = (int*)(ws + o);
    size_t oi = 0;
    int* perm      = wsI + oi; oi += (size_t)N_EDGES;        // 0.8M
    int* blockHist = wsI + oi; oi += (size_t)NBLK_E * RR;
    int* blockOfs  = wsI + oi; oi += (size_t)NBLK_E * RR;
    int* base      = wsI + oi; oi += RR + 1;
    int* tileBase  = wsI + oi; oi += RR + 1;
    // total ~71 MB

    float* out_x    = (float*)d_out;
    float* out_rel  = out_x + (size_t)N_GRAPHS * DD;
    float* out_edge = out_rel + (size_t)N_EDGES * HH;

    // initial features + relation sort (layer-invariant)
    argcn_gather_h<<<(N_NODES * DD + 255) / 256, 256, 0, stream>>>(code_emb, node_ids, h);
    argcn_sort_count<<<NBLK_E, 256, 0, stream>>>(rel, blockHist);
    argcn_sort_scan<<<1, 256, 0, stream>>>(blockHist, base, tileBase, blockOfs);
    argcn_sort_scatter<<<NBLK_E, 256, 0, stream>>>(rel, blockOfs, perm);

    for (int l = 0; l < LL; l++) {
        const float* Wr_l  = Wr + (size_t)l * DD * HH * DD;
        const float* br_l  = br + (size_t)l * HH * DD;
        const float* We_l  = We + (size_t)l * DD * HH * DD;
        const float* be_l  = be + (size_t)l * HH * DD;
        const float* W_l   = W  + (size_t)l * RR * DD * DD;
        const float* ah_l  = attn_h  + (size_t)l * HH * DD;
        const float* at_l  = attn_t  + (size_t)l * HH * DD;
        const float* as_l  = attn_s  + (size_t)l * HH * DD;
        const float* ar_l  = attn_r  + (size_t)l * HH * DD;
        const float* ars_l = attn_rs + (size_t)l * HH * DD;
        const float* hb_l  = h_bias  + (size_t)l * DD;

        argcn_init_layer<<<(N_NODES * DD + 255) / 256, 256, 0, stream>>>(agg, den, emax);
        argcn_rel_er<<<1, 64, 0, stream>>>(rel_tab, Wr_l, br_l, ar_l, er);
        argcn_graph_att<<<N_GRAPHS, 128, 0, stream>>>(state, Wr_l, br_l, We_l, be_l,
                                                      ars_l, as_l, er, ra, esg);
        argcn_hp_gemm<<<N_NODES / 16, 256, 0, stream>>>(h, We_l, be_l, hp);
        argcn_node_att<<<(N_NODES * HH + 255) / 256, 256, 0, stream>>>(hp, ah_l, at_l,
                                                                       esg, n2g, a_nh, b_nh);
        argcn_edge_pass1<<<(N_EDGES * HH + 255) / 256, 256, 0, stream>>>(src, dst, a_nh,
                                                                         b_nh, ebuf, emax);
        argcn_edge_pass2<<<(N_EDGES * HH + 255) / 256, 256, 0, stream>>>(dst, emax, ebuf, den);
        argcn_edge_coeff<<<(N_EDGES * HH + 255) / 256, 256, 0, stream>>>(src, dst, rel, n2g,
                                                                         ebuf, den, ra,
                                                                         out_edge, out_rel,
                                                                         coeff);
        argcn_edge_msg_wmma<<<(TILES_MAX + 7) / 8, 256, 0, stream>>>(perm, base, tileBase,
                                                                     src, dst, coeff, hp,
                                                                     W_l, agg);
        argcn_update_h<<<(N_NODES * DD + 255) / 256, 256, 0, stream>>>(agg, hb_l, h);
    }

    argcn_readout<<<N_GRAPHS, 32, 0, stream>>>(h, rw, out_x);
}